// WaveletMoCo_53678501266081
// MI455X (gfx1250) — compile-verified
//
#include <hip/hip_runtime.h>
#include <math.h>

typedef float v2f __attribute__((ext_vector_type(2)));
typedef float v8f __attribute__((ext_vector_type(8)));

#define BATCH   64
#define CH      3
#define HALF    112
#define FEAT    37632          // 3*112*112
#define DIM     128
#define NK      65536
#define NLOG    65537          // 1 + NK
#define NLL     (BATCH*CH*HALF*HALF)   // 2408448
#define KSLICE  896
#define NSLICES 42             // 42*896 = 37632

// ---------------------------------------------------------------------------
// Haar DWT: one thread per output pixel; float2 loads (8B aligned), LL -> ws,
// LH/HL/HH -> d_out. Pure streaming pass.
// ---------------------------------------------------------------------------
__global__ __launch_bounds__(256) void dwt_kernel(
    const float* __restrict__ im,
    float* __restrict__ LL, float* __restrict__ LH,
    float* __restrict__ HL, float* __restrict__ HHo)
{
    size_t idx = (size_t)blockIdx.x * 256 + threadIdx.x;
    if (idx >= (size_t)NLL) return;
    int j  = (int)(idx % HALF);
    size_t t = idx / HALF;
    int i  = (int)(t % HALF);
    size_t bc = t / HALF;
    const float* p = im + bc * (size_t)(224 * 224) + (size_t)(2 * i) * 224 + 2 * j;
    float2 top = *(const float2*)(p);
    float2 bot = *(const float2*)(p + 224);
    float a = top.x, b = top.y, c = bot.x, d = bot.y;
    LL[idx]  = (a + b + c + d) * 0.5f;
    LH[idx]  = (a + b - c - d) * 0.5f;
    HL[idx]  = (a - b + c - d) * 0.5f;
    HHo[idx] = (a - b - c + d) * 0.5f;
}

// ---------------------------------------------------------------------------
// Zero the two 64x128 accumulators (contiguous in ws) and the labels.
// ---------------------------------------------------------------------------
__global__ __launch_bounds__(256) void init_kernel(float* __restrict__ acc,
                                                   float* __restrict__ labels)
{
    int idx = blockIdx.x * 256 + threadIdx.x;
    if (idx < 2 * BATCH * DIM) acc[idx] = 0.0f;
    if (idx < BATCH) labels[idx] = 0.0f;   // int32 zero == float 0.0 bit pattern
}

// ---------------------------------------------------------------------------
// Split-K encoder GEMM: acc(64x128) += A(64xFEAT) * W(FEATx128)
// W = f1*W1 + f2*W2 (W2 may be null). Grid (4 rowTiles, 8 colTiles, 42 kSlices),
// one wave32 per block, v_wmma_f32_16x16x4_f32 chain, atomic f32 reduction.
//
// A 16x4 f32 layout: lanes 0-15 hold M=0..15 {K0,K1}; lanes 16-31 {K2,K3}.
// B 4x16 f32 layout: lanes 0-15 hold N=0..15 rows {K0,K1}; lanes 16-31 {K2,K3}.
// C/D: VGPR v, lanes 0-15 -> M=v, lanes 16-31 -> M=v+8, N = lane&15.
// ---------------------------------------------------------------------------
__global__ __launch_bounds__(32) void enc_gemm_kernel(
    const float* __restrict__ A,
    const float* __restrict__ W1, const float* __restrict__ W2,
    float f1, float f2,
    float* __restrict__ acc)
{
    const int lane  = threadIdx.x;
    const int rt    = blockIdx.x;           // 0..3
    const int ct    = blockIdx.y;           // 0..7
    const int ks    = blockIdx.z;           // 0..41
    const int m     = lane & 15;
    const int khalf = (lane >> 4) << 1;     // 0 or 2
    const int col   = ct * 16 + m;
    const float* arow = A + (size_t)(rt * 16 + m) * FEAT;
    const int k0 = ks * KSLICE;

    v8f c = {0.f, 0.f, 0.f, 0.f, 0.f, 0.f, 0.f, 0.f};

    if (W2) {
        #pragma unroll 4
        for (int k = k0; k < k0 + KSLICE; k += 4) {
            v2f a = *(const v2f*)(arow + k + khalf);
            size_t w = (size_t)(k + khalf) * DIM + col;
            v2f b;
            b.x = f1 * W1[w]       + f2 * W2[w];
            b.y = f1 * W1[w + DIM] + f2 * W2[w + DIM];
            c = __builtin_amdgcn_wmma_f32_16x16x4_f32(false, a, false, b,
                                                      (short)0, c, false, false);
        }
    } else {
        #pragma unroll 4
        for (int k = k0; k < k0 + KSLICE; k += 4) {
            v2f a = *(const v2f*)(arow + k + khalf);
            size_t w = (size_t)(k + khalf) * DIM + col;
            v2f b;
            b.x = W1[w];
            b.y = W1[w + DIM];
            c = __builtin_amdgcn_wmma_f32_16x16x4_f32(false, a, false, b,
                                                      (short)0, c, false, false);
        }
    }

    const int mb = (lane >> 4) * 8;
    #pragma unroll
    for (int v = 0; v < 8; ++v) {
        int r = rt * 16 + mb + v;
        atomicAdd(&acc[r * DIM + col], c[v]);
    }
}

// ---------------------------------------------------------------------------
// Bias + L2-normalize encoder outputs for q and (momentum-blended) k.
// One block (128 threads) per batch row.
// ---------------------------------------------------------------------------
__global__ __launch_bounds__(128) void finalize_enc_kernel(
    const float* __restrict__ accq, const float* __restrict__ acck,
    const float* __restrict__ bq,   const float* __restrict__ bk,
    float* __restrict__ q1, float* __restrict__ kf)
{
    const int row = blockIdx.x, t = threadIdx.x;
    __shared__ float red[128];
    float vq = accq[row * DIM + t] + bq[t];
    float vk = acck[row * DIM + t] + (0.999f * bk[t] + 0.001f * bq[t]);

    red[t] = vq * vq; __syncthreads();
    for (int s = 64; s > 0; s >>= 1) { if (t < s) red[t] += red[t + s]; __syncthreads(); }
    float nq = fmaxf(sqrtf(red[0]), 1e-12f); __syncthreads();

    red[t] = vk * vk; __syncthreads();
    for (int s = 64; s > 0; s >>= 1) { if (t < s) red[t] += red[t + s]; __syncthreads(); }
    float nk = fmaxf(sqrtf(red[0]), 1e-12f);

    q1[row * DIM + t] = vq / nq;
    kf[row * DIM + t] = vk / nk;
}

// ---------------------------------------------------------------------------
// MLP head: q2 = l2norm(relu(q1 @ w_p1 + b1) @ w_p2 + b2). One block per row.
// 2 MFLOP total -> trivial; keep it on VALU.
// ---------------------------------------------------------------------------
__global__ __launch_bounds__(128) void mlp_kernel(
    const float* __restrict__ q1,
    const float* __restrict__ w1, const float* __restrict__ b1,
    const float* __restrict__ w2, const float* __restrict__ b2,
    float* __restrict__ q2)
{
    const int row = blockIdx.x, t = threadIdx.x;
    __shared__ float x[128];
    __shared__ float h[128];
    __shared__ float red[128];

    x[t] = q1[row * DIM + t]; __syncthreads();

    float s = b1[t];
    #pragma unroll 8
    for (int cc = 0; cc < DIM; ++cc) s += x[cc] * w1[cc * DIM + t];
    h[t] = fmaxf(s, 0.0f); __syncthreads();

    float s2 = b2[t];
    #pragma unroll 8
    for (int cc = 0; cc < DIM; ++cc) s2 += h[cc] * w2[cc * DIM + t];

    red[t] = s2 * s2; __syncthreads();
    for (int sft = 64; sft > 0; sft >>= 1) { if (t < sft) red[t] += red[t + sft]; __syncthreads(); }
    float n = fmaxf(sqrtf(red[0]), 1e-12f);

    q2[row * DIM + t] = s2 / n;
}

// ---------------------------------------------------------------------------
// l_pos: logits[row][0] = dot(q2[row], k[row]) / T
// ---------------------------------------------------------------------------
__global__ __launch_bounds__(128) void lpos_kernel(
    const float* __restrict__ q2, const float* __restrict__ kf,
    float* __restrict__ logits, float invT)
{
    const int row = blockIdx.x, t = threadIdx.x;
    __shared__ float red[128];
    red[t] = q2[row * DIM + t] * kf[row * DIM + t]; __syncthreads();
    for (int s = 64; s > 0; s >>= 1) { if (t < s) red[t] += red[t + s]; __syncthreads(); }
    if (t == 0) logits[(size_t)row * NLOG] = red[0] * invT;
}

// ---------------------------------------------------------------------------
// l_neg: logits[:, 1:] = (q2 @ queue) / T via WMMA. Grid (4, 4096) wave32
// tiles, K = 128 -> 32 WMMAs per tile. queue streamed once, coalesced in N.
// ---------------------------------------------------------------------------
__global__ __launch_bounds__(32) void lneg_kernel(
    const float* __restrict__ q2, const float* __restrict__ queue,
    float* __restrict__ logits, float invT)
{
    const int lane  = threadIdx.x;
    const int rt    = blockIdx.x;          // 0..3
    const int ct    = blockIdx.y;          // 0..4095
    const int m     = lane & 15;
    const int khalf = (lane >> 4) << 1;
    const int col   = ct * 16 + m;
    const float* qrow = q2 + (size_t)(rt * 16 + m) * DIM;

    v8f c = {0.f, 0.f, 0.f, 0.f, 0.f, 0.f, 0.f, 0.f};
    #pragma unroll 8
    for (int k = 0; k < DIM; k += 4) {
        v2f a = *(const v2f*)(qrow + k + khalf);
        size_t w = (size_t)(k + khalf) * NK + col;
        v2f b;
        b.x = queue[w];
        b.y = queue[w + NK];
        c = __builtin_amdgcn_wmma_f32_16x16x4_f32(false, a, false, b,
                                                  (short)0, c, false, false);
    }

    const int mb = (lane >> 4) * 8;
    #pragma unroll
    for (int v = 0; v < 8; ++v) {
        int r = rt * 16 + mb + v;
        logits[(size_t)r * NLOG + 1 + col] = c[v] * invT;
    }
}

// ---------------------------------------------------------------------------
extern "C" void kernel_launch(void* const* d_in, const int* in_sizes, int n_in,
                              void* d_out, int out_size, void* d_ws, size_t ws_size,
                              hipStream_t stream)
{
    const float* im_q    = (const float*)d_in[0];
    const float* im_k    = (const float*)d_in[1];
    const float* w_enc_q = (const float*)d_in[2];
    const float* b_enc_q = (const float*)d_in[3];
    const float* w_enc_k = (const float*)d_in[4];
    const float* b_enc_k = (const float*)d_in[5];
    const float* w_p1    = (const float*)d_in[6];
    const float* b_p1    = (const float*)d_in[7];
    const float* w_p2    = (const float*)d_in[8];
    const float* b_p2    = (const float*)d_in[9];
    const float* queue   = (const float*)d_in[10];

    // d_out layout: logits (64*65537) | labels (64) | qLH | qHL | qHH | kLH | kHL | kHH
    float* out    = (float*)d_out;
    float* logits = out;
    float* labels = out + (size_t)BATCH * NLOG;
    float* qLH = labels + BATCH;
    float* qHL = qLH + NLL;
    float* qHH = qHL + NLL;
    float* kLH = qHH + NLL;
    float* kHL = kLH + NLL;
    float* kHH = kHL + NLL;

    // workspace: qLL | kLL | accq | acck | q1 | kf | q2  (~19.5 MB)
    float* ws   = (float*)d_ws;
    float* qLL  = ws;
    float* kLL  = qLL + NLL;
    float* accq = kLL + NLL;
    float* acck = accq + BATCH * DIM;
    float* q1   = acck + BATCH * DIM;
    float* kf   = q1 + BATCH * DIM;
    float* q2   = kf + BATCH * DIM;

    const float invT = 1.0f / 0.07f;

    dim3 dwtg((NLL + 255) / 256);
    dwt_kernel<<<dwtg, 256, 0, stream>>>(im_q, qLL, qLH, qHL, qHH);
    dwt_kernel<<<dwtg, 256, 0, stream>>>(im_k, kLL, kLH, kHL, kHH);

    init_kernel<<<(2 * BATCH * DIM + 255) / 256, 256, 0, stream>>>(accq, labels);

    enc_gemm_kernel<<<dim3(4, 8, NSLICES), 32, 0, stream>>>(
        qLL, w_enc_q, nullptr, 1.0f, 0.0f, accq);
    enc_gemm_kernel<<<dim3(4, 8, NSLICES), 32, 0, stream>>>(
        kLL, w_enc_k, w_enc_q, 0.999f, 0.001f, acck);

    finalize_enc_kernel<<<BATCH, 128, 0, stream>>>(accq, acck, b_enc_q, b_enc_k, q1, kf);
    mlp_kernel<<<BATCH, 128, 0, stream>>>(q1, w_p1, b_p1, w_p2, b_p2, q2);

    lpos_kernel<<<BATCH, 128, 0, stream>>>(q2, kf, logits, invT);
    lneg_kernel<<<dim3(4, NK / 16), 32, 0, stream>>>(q2, queue, logits, invT);
}